// PCLEmbeddings_85083302134221
// MI455X (gfx1250) — compile-verified
//
#include <hip/hip_runtime.h>
#include <hip/hip_bf16.h>

// Problem dims (fixed by the reference)
#define B_  32
#define S_  512
#define H_  1024
#define P_  16
#define LN_EPS 1e-5f

typedef __attribute__((ext_vector_type(2))) float v2f;
typedef __attribute__((ext_vector_type(8))) float v8f;

// ---------------------------------------------------------------------------
// Kernel 1/2: prompt-encoder MLP layer via V_WMMA_F32_16X16X4_F32.
// Computes OUT[0:16, :] = act( X[16,K=H] @ W[N=H,K=H]^T + bias ), torch-Linear
// convention (out[p,n] = sum_k X[p,k] * W[n,k]).
// One wave (32 lanes) owns one 16x16 output tile; 64 tiles cover N=1024.
// Grid = 8 blocks x 256 threads (8 waves) -> exactly 64 full waves, EXEC all-1s.
// ---------------------------------------------------------------------------
__global__ __launch_bounds__(256) void prompt_mlp_wmma(
    const float* __restrict__ X,     // [16, H]
    const float* __restrict__ W,     // [H, H] row-major (torch Linear weight)
    const float* __restrict__ bias,  // [H]
    float*       __restrict__ OUT,   // [16, H]
    int do_relu)
{
  const int tile = (blockIdx.x * blockDim.x + threadIdx.x) >> 5; // 0..63
  const int lane = threadIdx.x & 31;
  const int rn   = lane & 15;          // A: M row / B: N column
  const int kh   = (lane >> 4) << 1;   // K sub-offset {0,2} within 4-wide step

  const float* arow = X + (size_t)rn * H_;                  // A row m
  const float* brow = W + (size_t)(tile * 16 + rn) * H_;    // B col n == W row n

  v8f acc = {0.f, 0.f, 0.f, 0.f, 0.f, 0.f, 0.f, 0.f};

#pragma unroll 4
  for (int k0 = 0; k0 < H_; k0 += 4) {
    // 32-bit A 16x4 layout: lanes 0-15 -> K={k0,k0+1}, lanes 16-31 -> K={k0+2,k0+3}
    v2f a = *(const v2f*)(arow + k0 + kh);
    v2f b = *(const v2f*)(brow + k0 + kh);
    acc = __builtin_amdgcn_wmma_f32_16x16x4_f32(
        /*neg_a=*/false, a, /*neg_b=*/false, b,
        /*c_mod=*/(short)0, acc, /*reuse_a=*/false, /*reuse_b=*/false);
  }

  // C/D 16x16 f32 layout: n = lane&15, m = 8*(lane>>4) + vgpr
  const int   n     = tile * 16 + rn;
  const float bv    = bias[n];
  const int   mbase = (lane >> 4) << 3;
#pragma unroll
  for (int v = 0; v < 8; ++v) {
    float r = acc[v] + bv;
    if (do_relu) r = fmaxf(r, 0.f);
    OUT[(size_t)(mbase + v) * H_ + n] = r;
  }
}

// ---------------------------------------------------------------------------
// Kernel 3: embedding gather + prompt scatter + pos/token add + LayerNorm.
// One wave32 per (b,s) row: 32 lanes x 8 float4 = 1024 = H elements held in
// registers; single pass (load -> wave-reduce mean/var via shfl -> write).
// Memory-bound: ~128 MB total traffic -> ~5.6 us at 23.3 TB/s.
// ---------------------------------------------------------------------------
__global__ __launch_bounds__(256) void embed_ln_kernel(
    const int*   __restrict__ input_ids,     // [B, S]
    const int*   __restrict__ prompt_pos,    // [P]
    const float* __restrict__ W_word,        // [VOCAB, H]
    const float* __restrict__ W_pos,         // [MAXPOS, H]
    const float* __restrict__ W_tok,         // [1, H]
    const float* __restrict__ prompt_embeds, // [P, H]
    const float* __restrict__ gamma,         // [H]
    const float* __restrict__ beta,          // [H]
    float*       __restrict__ out)           // [B, S, H]
{
  const int lane = threadIdx.x & 31;
  const int wib  = threadIdx.x >> 5;
  const int row  = blockIdx.x * (blockDim.x >> 5) + wib;   // 0 .. B*S-1
  const int s    = row & (S_ - 1);

  // prompt scatter: last matching index wins (scatter-set semantics)
  int pidx = -1;
#pragma unroll
  for (int p = 0; p < P_; ++p)
    if (prompt_pos[p] == s) pidx = p;

  const float* src = (pidx >= 0)
      ? (prompt_embeds + (size_t)pidx * H_)
      : (W_word + (size_t)input_ids[row] * H_);
  const float* pos = W_pos + (size_t)(s + 2) * H_;   // position id = s + PAD_IDX + 1

  float4 v[8];
  float sum = 0.f;
#pragma unroll
  for (int j = 0; j < 8; ++j) {
    const int idx = lane * 4 + j * 128;  // 32 lanes x 16B contiguous per step
    float4 a  = *(const float4*)(src   + idx);
    float4 p4 = *(const float4*)(pos   + idx);
    float4 t4 = *(const float4*)(W_tok + idx);
    a.x += p4.x + t4.x;
    a.y += p4.y + t4.y;
    a.z += p4.z + t4.z;
    a.w += p4.w + t4.w;
    v[j] = a;
    sum += a.x + a.y + a.z + a.w;
  }

  // wave32 reduction: mean
#pragma unroll
  for (int off = 16; off > 0; off >>= 1) sum += __shfl_xor(sum, off, 32);
  const float mean = sum * (1.f / H_);

  float var = 0.f;
#pragma unroll
  for (int j = 0; j < 8; ++j) {
    float dx = v[j].x - mean, dy = v[j].y - mean,
          dz = v[j].z - mean, dw = v[j].w - mean;
    var += dx * dx + dy * dy + dz * dz + dw * dw;
  }
#pragma unroll
  for (int off = 16; off > 0; off >>= 1) var += __shfl_xor(var, off, 32);
  const float rstd = rsqrtf(var * (1.f / H_) + LN_EPS);

  float* orow = out + (size_t)row * H_;
#pragma unroll
  for (int j = 0; j < 8; ++j) {
    const int idx = lane * 4 + j * 128;
    float4 g = *(const float4*)(gamma + idx);
    float4 bt = *(const float4*)(beta + idx);
    float4 o;
    o.x = g.x * (v[j].x - mean) * rstd + bt.x;
    o.y = g.y * (v[j].y - mean) * rstd + bt.y;
    o.z = g.z * (v[j].z - mean) * rstd + bt.z;
    o.w = g.w * (v[j].w - mean) * rstd + bt.w;
    *(float4*)(orow + idx) = o;
  }
}

// ---------------------------------------------------------------------------
// Launch
// ---------------------------------------------------------------------------
extern "C" void kernel_launch(void* const* d_in, const int* in_sizes, int n_in,
                              void* d_out, int out_size, void* d_ws, size_t ws_size,
                              hipStream_t stream) {
  const int*   input_ids    = (const int*)  d_in[0];
  const int*   prompt_pos   = (const int*)  d_in[1];
  const float* W_word       = (const float*)d_in[2];
  const float* W_pos        = (const float*)d_in[3];
  const float* W_tok        = (const float*)d_in[4];
  const float* prompt_table = (const float*)d_in[5];
  const float* W1           = (const float*)d_in[6];
  const float* b1           = (const float*)d_in[7];
  const float* W2           = (const float*)d_in[8];
  const float* b2           = (const float*)d_in[9];
  const float* gamma        = (const float*)d_in[10];
  const float* beta         = (const float*)d_in[11];
  float*       out          = (float*)d_out;

  // scratch: h [16,1024] then prompt_embeds [16,1024]
  float* h_buf  = (float*)d_ws;
  float* pe_buf = h_buf + P_ * H_;

  // MLP layer 1: h = relu(prompt_table @ W1^T + b1)   (64 tiles = 8 blocks x 8 waves)
  prompt_mlp_wmma<<<8, 256, 0, stream>>>(prompt_table, W1, b1, h_buf, /*relu=*/1);
  // MLP layer 2: prompt_embeds = h @ W2^T + b2
  prompt_mlp_wmma<<<8, 256, 0, stream>>>(h_buf, W2, b2, pe_buf, /*relu=*/0);

  // Fused gather + scatter + add + LayerNorm: one wave per row, 8 waves/block
  const int rows = B_ * S_;                 // 16384
  const int blocks = rows / 8;              // 2048
  embed_ln_kernel<<<blocks, 256, 0, stream>>>(
      input_ids, prompt_pos, W_word, W_pos, W_tok, pe_buf, gamma, beta, out);
}